// ResConv_1133871366243
// MI455X (gfx1250) — compile-verified
//
#include <hip/hip_runtime.h>

#define NN 100000
#define NE 1250000
#define D  64

typedef float v2f __attribute__((ext_vector_type(2)));
typedef float v8f __attribute__((ext_vector_type(8)));

__device__ __forceinline__ v8f wmma4(v2f a, v2f b, v8f c) {
  // V_WMMA_F32_16X16X4_F32 : D = A(16x4) x B(4x16) + C(16x16), all f32
  return __builtin_amdgcn_wmma_f32_16x16x4_f32(
      /*neg_a=*/false, a, /*neg_b=*/false, b,
      /*c_mod=*/(short)0, c, /*reuse_a=*/false, /*reuse_b=*/false);
}

__device__ __forceinline__ void atomic_add_f32(float* p, float v) {
  unsafeAtomicAdd(p, v);  // hardware global_atomic_add_f32
}

// ---------------- degree / normalization ----------------
__global__ void count_deg_kernel(const int* __restrict__ dst, float* __restrict__ deg, int E) {
  int e = blockIdx.x * blockDim.x + threadIdx.x;
  if (e < E) atomic_add_f32(&deg[dst[e]], 1.0f);
}

__global__ void finalize_deg_kernel(const float* __restrict__ deg, float* __restrict__ dis,
                                    float* __restrict__ invdeg, int n) {
  int i = blockIdx.x * blockDim.x + threadIdx.x;
  if (i < n) {
    float d = deg[i];
    dis[i]    = rsqrtf(d + 1.0f);       // GCN: deg includes self loop
    invdeg[i] = 1.0f / fmaxf(d, 1.0f);  // SAGE mean
  }
}

// ---------------- elementwise ----------------
// out[i,:] = in[i,:] * dis[i]^2   (GCN self-loop message; also initializes agg buffer)
__global__ void self_loop_kernel(const float* __restrict__ in, const float* __restrict__ dis,
                                 float* __restrict__ out, int n) {
  int t = blockIdx.x * blockDim.x + threadIdx.x;
  int i = t >> 4;
  int f = (t & 15) << 2;
  if (i < n) {
    float s = dis[i]; s = s * s;
    float4 v = *(const float4*)(in + (size_t)i * D + f);
    v.x *= s; v.y *= s; v.z *= s; v.w *= s;
    *(float4*)(out + (size_t)i * D + f) = v;
  }
}

// out[i,:] = in[i,:] + bias[:]  (+relu)
__global__ void bias_act_kernel(const float* __restrict__ in, const float* __restrict__ bias,
                                float* __restrict__ out, int n, int relu) {
  int t = blockIdx.x * blockDim.x + threadIdx.x;
  int i = t >> 4;
  int f = (t & 15) << 2;
  if (i < n) {
    float4 v = *(const float4*)(in + (size_t)i * D + f);
    const float4 b = *(const float4*)(bias + f);
    v.x += b.x; v.y += b.y; v.z += b.z; v.w += b.w;
    if (relu) {
      v.x = fmaxf(v.x, 0.0f); v.y = fmaxf(v.y, 0.0f);
      v.z = fmaxf(v.z, 0.0f); v.w = fmaxf(v.w, 0.0f);
    }
    *(float4*)(out + (size_t)i * D + f) = v;
  }
}

// ---------------- edge scatter ----------------
// one wave per edge, each lane covers 2 features (64 total)
__global__ void gcn_edge_kernel(const float* __restrict__ xw, const float* __restrict__ dis,
                                const int* __restrict__ src, const int* __restrict__ dst,
                                float* __restrict__ agg, int E) {
  int t = blockIdx.x * blockDim.x + threadIdx.x;
  int e = t >> 5;
  int lane = t & 31;
  if (e < E) {
    int s = src[e], d = dst[e];
    float w = dis[s] * dis[d];
    v2f v = *(const v2f*)(xw + (size_t)s * D + lane * 2);
    atomic_add_f32(&agg[(size_t)d * D + lane * 2 + 0], v.x * w);
    atomic_add_f32(&agg[(size_t)d * D + lane * 2 + 1], v.y * w);
  }
}

__global__ void edge_sum_kernel(const float* __restrict__ h,
                                const int* __restrict__ src, const int* __restrict__ dst,
                                float* __restrict__ sum, int E) {
  int t = blockIdx.x * blockDim.x + threadIdx.x;
  int e = t >> 5;
  int lane = t & 31;
  if (e < E) {
    int s = src[e], d = dst[e];
    v2f v = *(const v2f*)(h + (size_t)s * D + lane * 2);
    atomic_add_f32(&sum[(size_t)d * D + lane * 2 + 0], v.x);
    atomic_add_f32(&sum[(size_t)d * D + lane * 2 + 1], v.y);
  }
}

// ---------------- WMMA GEMM ----------------
// out = act( (A1 .* rs1) @ W1 [+ A2 @ W2] [+ bias] [+ resid] )
// A: [n,64] row-major, W: [64,64] row-major, one wave per 16-row tile.
__global__ __launch_bounds__(256) void gemm64_kernel(
    const float* __restrict__ A1, const float* __restrict__ rs1,
    const float* __restrict__ W1,
    const float* __restrict__ A2, const float* __restrict__ W2,
    const float* __restrict__ bias, const float* __restrict__ resid,
    float* __restrict__ out, int n, int relu)
{
  const int gt   = blockIdx.x * blockDim.x + threadIdx.x;
  const int wave = gt >> 5;
  const int lane = threadIdx.x & 31;
  const int row0 = wave << 4;
  if (row0 >= n) return;  // wave-uniform: EXEC stays all-ones for WMMA

  const int m  = lane & 15;   // A: row M ; B/C/D: column N (mod 16)
  const int kh = lane >> 4;   // K half (A/B) ; M half (C/D)
  const int ra = row0 + m;

  // A fragments: lane holds A[ra][kb*4 + kh*2 + {0,1}] per the 16x4 f32 layout
  v2f a1[16];
  {
    const float* ap = A1 + (size_t)ra * D + kh * 2;
    const float s = rs1 ? rs1[ra] : 1.0f;
#pragma unroll
    for (int kb = 0; kb < 16; ++kb) {
      v2f t = *(const v2f*)(ap + kb * 4);
      a1[kb] = t * s;
    }
  }
  v2f a2[16];
  if (A2) {
    const float* ap = A2 + (size_t)ra * D + kh * 2;
#pragma unroll
    for (int kb = 0; kb < 16; ++kb) a2[kb] = *(const v2f*)(ap + kb * 4);
  }

  v8f acc[4] = {};

#pragma unroll
  for (int t = 0; t < 4; ++t) {
    const int nc = t * 16 + m;
#pragma unroll
    for (int kb = 0; kb < 16; ++kb) {
      const int k = kb * 4 + kh * 2;
      v2f b;
      b.x = W1[k * D + nc];
      b.y = W1[k * D + D + nc];
      acc[t] = wmma4(a1[kb], b, acc[t]);
    }
  }
  if (A2) {
#pragma unroll
    for (int t = 0; t < 4; ++t) {
      const int nc = t * 16 + m;
#pragma unroll
      for (int kb = 0; kb < 16; ++kb) {
        const int k = kb * 4 + kh * 2;
        v2f b;
        b.x = W2[k * D + nc];
        b.y = W2[k * D + D + nc];
        acc[t] = wmma4(a2[kb], b, acc[t]);
      }
    }
  }

  // C/D layout: VGPR r holds row (row0 + r + 8*kh), column nc
#pragma unroll
  for (int t = 0; t < 4; ++t) {
    const int nc = t * 16 + m;
    const float bv = bias ? bias[nc] : 0.0f;
#pragma unroll
    for (int r = 0; r < 8; ++r) {
      const int row = row0 + r + 8 * kh;
      float v = acc[t][r] + bv;
      if (resid) v += resid[(size_t)row * D + nc];
      if (relu)  v = fmaxf(v, 0.0f);
      out[(size_t)row * D + nc] = v;
    }
  }
}

// ---------------- host ----------------
extern "C" void kernel_launch(void* const* d_in, const int* in_sizes, int n_in,
                              void* d_out, int out_size, void* d_ws, size_t ws_size,
                              hipStream_t stream) {
  (void)in_sizes; (void)n_in; (void)out_size; (void)ws_size;
  const float* x        = (const float*)d_in[0];
  const int*   eidx     = (const int*)d_in[1];
  const int*   src      = eidx;
  const int*   dst      = eidx + NE;
  const float* gcn1_w   = (const float*)d_in[2];
  const float* gcn1_b   = (const float*)d_in[3];
  const float* sage1_lw = (const float*)d_in[4];
  const float* sage1_lb = (const float*)d_in[5];
  const float* sage1_rw = (const float*)d_in[6];
  const float* sage2_lw = (const float*)d_in[7];
  const float* sage2_lb = (const float*)d_in[8];
  const float* sage2_rw = (const float*)d_in[9];
  const float* lin_w    = (const float*)d_in[10];
  const float* lin_b    = (const float*)d_in[11];
  const float* gcn2_w   = (const float*)d_in[12];
  const float* gcn2_b   = (const float*)d_in[13];
  float* out = (float*)d_out;

  float* ws     = (float*)d_ws;
  float* deg    = ws;
  float* dis    = ws + NN;
  float* invdeg = ws + 2 * NN;
  float* b0     = ws + 3 * NN;
  float* b1     = b0 + (size_t)NN * D;
  float* b2     = b1 + (size_t)NN * D;

  const int TB = 256;
  auto cdiv = [](long a, long b) { return (int)((a + b - 1) / b); };
  const int eg = cdiv((long)NE * 32, TB);     // edge kernels: one wave per edge
  const int ng = cdiv((long)NN * 16, TB);     // elementwise: float4 per thread
  const int gg = cdiv(((long)NN / 16) * 32, TB);  // gemm: one wave per 16 rows
  const size_t nb = (size_t)NN * D * sizeof(float);

  // degrees + normalizations (shared by both GCN layers and SAGE means)
  hipMemsetAsync(deg, 0, NN * sizeof(float), stream);
  count_deg_kernel<<<cdiv(NE, TB), TB, 0, stream>>>(dst, deg, NE);
  finalize_deg_kernel<<<cdiv(NN, TB), TB, 0, stream>>>(deg, dis, invdeg, NN);

  // layer 0: GCN1 -> relu      h -> b0
  gemm64_kernel<<<gg, TB, 0, stream>>>(x, nullptr, gcn1_w, nullptr, nullptr,
                                       nullptr, nullptr, b0, NN, 0);       // xw -> b0
  self_loop_kernel<<<ng, TB, 0, stream>>>(b0, dis, b1, NN);                // agg init -> b1
  gcn_edge_kernel<<<eg, TB, 0, stream>>>(b0, dis, src, dst, b1, NE);       // scatter
  bias_act_kernel<<<ng, TB, 0, stream>>>(b1, gcn1_b, b0, NN, 1);           // h -> b0

  // layer 1: SAGE1 + residual -> relu      h2 -> b1
  hipMemsetAsync(b2, 0, nb, stream);
  edge_sum_kernel<<<eg, TB, 0, stream>>>(b0, src, dst, b2, NE);
  gemm64_kernel<<<gg, TB, 0, stream>>>(b2, invdeg, sage1_lw, b0, sage1_rw,
                                       sage1_lb, b0, b1, NN, 1);

  // layer 2: SAGE2 + residual (no relu)    -> b0
  hipMemsetAsync(b2, 0, nb, stream);
  edge_sum_kernel<<<eg, TB, 0, stream>>>(b1, src, dst, b2, NE);
  gemm64_kernel<<<gg, TB, 0, stream>>>(b2, invdeg, sage2_lw, b1, sage2_rw,
                                       sage2_lb, b1, b0, NN, 0);

  // lin_resGCN -> relu                     h3 -> b2
  gemm64_kernel<<<gg, TB, 0, stream>>>(b0, nullptr, lin_w, nullptr, nullptr,
                                       lin_b, nullptr, b2, NN, 1);

  // layer 3: GCN2 (no relu) -> out
  gemm64_kernel<<<gg, TB, 0, stream>>>(b2, nullptr, gcn2_w, nullptr, nullptr,
                                       nullptr, nullptr, b0, NN, 0);       // xw2 -> b0
  self_loop_kernel<<<ng, TB, 0, stream>>>(b0, dis, b1, NN);
  gcn_edge_kernel<<<eg, TB, 0, stream>>>(b0, dis, src, dst, b1, NE);
  bias_act_kernel<<<ng, TB, 0, stream>>>(b1, gcn2_b, out, NN, 0);
}